// CenterLoss_48979807044131
// MI455X (gfx1250) — compile-verified
//
#include <hip/hip_runtime.h>

// CenterLoss: sum((x - centers[labels])^2) over (65536 x 512) f32.
// Bandwidth-bound: ~134 MB traffic / 23.3 TB/s  =>  ~5.8 us floor.
// Strategy: stream x with grouped b128 loads (8 in flight per wave per row),
// gather centers rows (L2-resident, 1 MiB), VALU fmaf for diff^2, then a
// WMMA f32 16x16x4 (B = ones) reduction tree + ds_bpermute finisher.

typedef float v2f __attribute__((ext_vector_type(2)));
typedef float v8f __attribute__((ext_vector_type(8)));

// Reduce 256 floats in LDS (p) to a single scalar using a chain of 4
// accumulating V_WMMA_F32_16X16X4_F32 ops with an all-ones B matrix:
//   D[m,n] += sum_k A[m,k]   (exact f32 row-sums)
// A-tile layout (16x4 f32, wave32): lanes 0-15 hold K=0,1 in v[0],v[1];
// lanes 16-31 hold K=2,3. C/D layout: lane n (0-15) VGPR r = D[r, n],
// lane n+16 VGPR r = D[r+8, n]. So total = sum8(lane j) + sum8(lane j+16).
// Must be called by a full wave (EXEC all ones). Result valid on lane 0.
__device__ __forceinline__ float wmma_reduce_256(const float* __restrict__ p,
                                                 int lane) {
  v8f c = {};
  const v2f bones = {1.0f, 1.0f};
  const int m  = lane & 15;
  const int kb = (lane >> 4) << 1;  // 0 for lanes 0-15, 2 for lanes 16-31
#pragma unroll
  for (int ch = 0; ch < 4; ++ch) {
    v2f a;
    a[0] = p[ch * 64 + m * 4 + kb + 0];
    a[1] = p[ch * 64 + m * 4 + kb + 1];
    // 8 args: (neg_a, A, neg_b, B, c_mod, C, reuse_a, reuse_b)
    c = __builtin_amdgcn_wmma_f32_16x16x4_f32(
        /*neg_a=*/false, a, /*neg_b=*/false, bones,
        /*c_mod=*/(short)0, c, /*reuse_a=*/false, /*reuse_b=*/false);
  }
  float s = c[0] + c[1] + c[2] + c[3] + c[4] + c[5] + c[6] + c[7];
  s += __shfl_xor(s, 16, 32);  // combine M=0..7 half with M=8..15 half
  return s;
}

// Specialized streaming kernel: DQ = D/4 float4s per row, DQ % 32 == 0.
// Per row, each lane issues DQ/32 x-loads and DQ/32 centers-loads grouped
// back-to-back (b128 each) before any wait -> deep per-wave MLP.
template <int DQ>
__global__ void __launch_bounds__(256)
center_loss_partial_fixed(const float* __restrict__ x,
                          const int* __restrict__ labels,
                          const float* __restrict__ centers,
                          float* __restrict__ partials,
                          int B) {
  constexpr int D  = DQ * 4;
  constexpr int NI = DQ / 32;  // float4 chunks per lane per row (4 for D=512)
  __shared__ float p[256];

  const int lane = threadIdx.x & 31;
  const int wib  = threadIdx.x >> 5;          // wave in block
  const int wpb  = blockDim.x >> 5;           // waves per block (8)
  const int gw   = blockIdx.x * wpb + wib;    // global wave id
  const int nw   = gridDim.x * wpb;           // total waves

  float4 acc = make_float4(0.f, 0.f, 0.f, 0.f);

  for (int row = gw; row < B; row += nw) {
    const int l = labels[row];                // wave-uniform -> scalar load
    const float4* __restrict__ xr =
        (const float4*)(x + (size_t)row * (size_t)D);
    const float4* __restrict__ cr =
        (const float4*)(centers + (size_t)l * (size_t)D);

    // Prefetch this wave's next row of x (one row ahead ~ right HBM
    // distance at roofline speed; speculative, dropped if OOB).
    __builtin_prefetch(x + ((size_t)row + (size_t)nw) * (size_t)D + lane * 16,
                       0, 0);

    float4 xv[NI], cv[NI];
#pragma unroll
    for (int it = 0; it < NI; ++it) xv[it] = xr[lane + it * 32];
#pragma unroll
    for (int it = 0; it < NI; ++it) cv[it] = cr[lane + it * 32];
#pragma unroll
    for (int it = 0; it < NI; ++it) {
      float dx = xv[it].x - cv[it].x;
      float dy = xv[it].y - cv[it].y;
      float dz = xv[it].z - cv[it].z;
      float dw = xv[it].w - cv[it].w;
      acc.x = fmaf(dx, dx, acc.x);
      acc.y = fmaf(dy, dy, acc.y);
      acc.z = fmaf(dz, dz, acc.z);
      acc.w = fmaf(dw, dw, acc.w);
    }
  }

  p[threadIdx.x] = (acc.x + acc.y) + (acc.z + acc.w);
  __syncthreads();

  if (wib == 0) {
    float s = wmma_reduce_256(p, lane);
    if (lane == 0) partials[blockIdx.x] = s;
  }
}

// Generic fallback for arbitrary D (multiple of 4).
__global__ void __launch_bounds__(256)
center_loss_partial_generic(const float* __restrict__ x,
                            const int* __restrict__ labels,
                            const float* __restrict__ centers,
                            float* __restrict__ partials,
                            int B, int D) {
  __shared__ float p[256];
  const int lane = threadIdx.x & 31;
  const int wib  = threadIdx.x >> 5;
  const int wpb  = blockDim.x >> 5;
  const int gw   = blockIdx.x * wpb + wib;
  const int nw   = gridDim.x * wpb;
  const int Dq   = D >> 2;

  float4 acc = make_float4(0.f, 0.f, 0.f, 0.f);
  for (int row = gw; row < B; row += nw) {
    const int l = labels[row];
    const float4* __restrict__ xr =
        (const float4*)(x + (size_t)row * (size_t)D);
    const float4* __restrict__ cr =
        (const float4*)(centers + (size_t)l * (size_t)D);
    for (int i = lane; i < Dq; i += 32) {
      float4 xv = xr[i];
      float4 cv = cr[i];
      float dx = xv.x - cv.x;
      float dy = xv.y - cv.y;
      float dz = xv.z - cv.z;
      float dw = xv.w - cv.w;
      acc.x = fmaf(dx, dx, acc.x);
      acc.y = fmaf(dy, dy, acc.y);
      acc.z = fmaf(dz, dz, acc.z);
      acc.w = fmaf(dw, dw, acc.w);
    }
  }

  p[threadIdx.x] = (acc.x + acc.y) + (acc.z + acc.w);
  __syncthreads();
  if (wib == 0) {
    float s = wmma_reduce_256(p, lane);
    if (lane == 0) partials[blockIdx.x] = s;
  }
}

__global__ void __launch_bounds__(256)
final_reduce(const float* __restrict__ partials, int n,
             float* __restrict__ out) {
  __shared__ float p[256];
  float s = 0.f;
  for (int i = threadIdx.x; i < n; i += 256) s += partials[i];
  p[threadIdx.x] = s;
  __syncthreads();
  if ((threadIdx.x >> 5) == 0) {
    float t = wmma_reduce_256(p, threadIdx.x & 31);
    if ((threadIdx.x & 31) == 0) out[0] = t;
  }
}

extern "C" void kernel_launch(void* const* d_in, const int* in_sizes, int n_in,
                              void* d_out, int out_size, void* d_ws,
                              size_t ws_size, hipStream_t stream) {
  const float* x        = (const float*)d_in[0];
  const int*   labels   = (const int*)d_in[1];
  const float* centers  = (const float*)d_in[2];
  float*       out      = (float*)d_out;
  float*       partials = (float*)d_ws;

  const int B = in_sizes[1];                       // 65536
  const int D = (B > 0) ? (in_sizes[0] / B) : 0;   // 512

  int nblocks = 2048;                              // 16384 wave32s in flight
  size_t need = (size_t)nblocks * sizeof(float);
  if (need > ws_size) nblocks = (int)(ws_size / sizeof(float));
  if (nblocks < 1) nblocks = 1;

  if (D == 512) {
    center_loss_partial_fixed<128><<<nblocks, 256, 0, stream>>>(
        x, labels, centers, partials, B);
  } else {
    center_loss_partial_generic<<<nblocks, 256, 0, stream>>>(
        x, labels, centers, partials, B, D);
  }
  final_reduce<<<1, 256, 0, stream>>>(partials, nblocks, out);
}